// LocalContextNorm_79096117723689
// MI455X (gfx1250) — compile-verified
//
#include <hip/hip_runtime.h>

typedef __attribute__((ext_vector_type(2))) float    v2f;
typedef __attribute__((ext_vector_type(8))) float    v8f;
typedef __attribute__((ext_vector_type(4))) unsigned v4u;
typedef __attribute__((ext_vector_type(4))) int      v4i;
typedef __attribute__((ext_vector_type(8))) int      v8i;

#define HH    512
#define WIDTH 512
#define CC    32
#define TILE  64
#define CST   81   // column-sum row stride (odd -> conflict-free strided WMMA A-reads)

// Issue one TDM 2-D tile load: global (rowstride 512 f32) -> LDS (packed rows).
__device__ __forceinline__ void tdm_load_tile(const float* gbase, float* ldst,
                                              int ph, int pw)
{
    unsigned long long ga = (unsigned long long)(uintptr_t)gbase;
    unsigned lds = (unsigned)(uintptr_t)ldst;   // low 32 bits of generic ptr = LDS offset

    v4u g0;
    g0[0] = 1u;                                  // count=1, no gather, user mode
    g0[1] = lds;                                 // lds_addr (bytes)
    g0[2] = (unsigned)ga;                        // global_addr[31:0]
    g0[3] = (unsigned)(ga >> 32) | 0x80000000u;  // global_addr[56:32] | type=2<<30

    v8i g1;
    g1[0] = 0x00020000;                          // data_size=2 (4B), no pad/iter/multicast
    g1[1] = (int)(512u << 16);                   // tensor_dim0[15:0] in [31:16]
    g1[2] = (int)(512u << 16);                   // tensor_dim0[31:16]=0 | tensor_dim1[15:0]
    g1[3] = (int)((unsigned)pw << 16);           // tensor_dim1[31:16]=0 | tile_dim0
    g1[4] = ph;                                  // tile_dim1 | tile_dim2=0
    g1[5] = 512;                                 // tensor_dim0_stride[31:0]
    g1[6] = 0;                                   // stride[47:32] | dim1_stride[15:0]
    g1[7] = 0;

    v4i z4 = {0, 0, 0, 0};
#if __clang_major__ >= 23
    v8i z8 = {0, 0, 0, 0, 0, 0, 0, 0};
    __builtin_amdgcn_tensor_load_to_lds(g0, g1, z4, z4, z8, 0);
#else
    __builtin_amdgcn_tensor_load_to_lds(g0, g1, z4, z4, 0);
#endif
}

__global__ __launch_bounds__(256)
void lcn_kernel(const float* __restrict__ x,
                const float* __restrict__ gamma,
                const float* __restrict__ beta,
                float* __restrict__ out)
{
    __shared__ float sX0[80 * 80];    // packed rows, runtime stride pw
    __shared__ float sX1[80 * 80];
    __shared__ float sCS[64 * CST];   // vertical 16-row window sums of (x0+x1)
    __shared__ float sCQ[64 * CST];   // vertical 16-row window sums of (x0^2+x1^2)

    const int tid = threadIdx.x;
    const int bz  = blockIdx.z;
    const int b   = bz >> 4;
    const int g   = bz & 15;
    const int c0  = g * 2, c1 = c0 + 1;
    const int y0  = blockIdx.y * TILE;
    const int x0  = blockIdx.x * TILE;

    // Window rows for output pixel p: clamp(p-8,0,495)+1 .. +16 (integral-image offset).
    const int wr0  = (y0 > 8 ? y0 - 8 : 0) + 1;
    const int wr63 = ((y0 + 55) < 495 ? (y0 + 55) : 495) + 1;
    const int pr0  = wr0 < y0 ? wr0 : y0;          // include output rows
    const int pr1  = wr63 + 15;
    const int ph   = pr1 - pr0 + 1;                // <= 79

    const int wc0  = (x0 > 8 ? x0 - 8 : 0) + 1;
    const int wc63 = ((x0 + 55) < 495 ? (x0 + 55) : 495) + 1;
    const int pc0  = wc0 < x0 ? wc0 : x0;
    const int pc1  = wc63 + 15;
    const int pw   = pc1 - pc0 + 1;                // <= 79

    const float ga0 = gamma[c0], ga1 = gamma[c1];
    const float be0 = beta[c0],  be1 = beta[c1];

    const float* xc0 = x + (size_t)(b * CC + c0) * HH * WIDTH;
    const float* xc1 = x + (size_t)(b * CC + c1) * HH * WIDTH;

    // ---------------- TDM: DMA both channel patches into LDS ----------------
    // TDM ignores EXEC -> must be reached by exactly one wave (scalar branch).
    if (__builtin_amdgcn_readfirstlane(tid) == 0) {
        const size_t tileoff = (size_t)pr0 * WIDTH + pc0;
        tdm_load_tile(xc0 + tileoff, sX0, ph, pw);
        tdm_load_tile(xc1 + tileoff, sX1, ph, pw);
        __builtin_amdgcn_s_wait_tensorcnt(0);
    }
    __syncthreads();

    // ---------------- vertical sliding 16-row window sums ----------------
    if (tid < 160) {
        int typ = tid / 80;            // 0:S  1:Q
        int k   = tid - typ * 80;
        float* dst = typ ? sCQ : sCS;
        if (k < pw) {
            int   cur = wr0 - pr0;     // 0 (interior) or 1 (top edge)
            float win = 0.0f;
            for (int r = 0; r < 16; ++r) {
                float a = sX0[(cur + r) * pw + k];
                float c = sX1[(cur + r) * pw + k];
                win += typ ? (a * a + c * c) : (a + c);
            }
            dst[k] = win;
            for (int ty = 1; ty < 64; ++ty) {
                int wg = y0 + ty - 8;
                wg = wg < 0 ? 0 : (wg > 495 ? 495 : wg);
                int p = wg + 1 - pr0;
                if (p != cur) {        // advance window by one row
                    float an = sX0[(cur + 16) * pw + k];
                    float cn = sX1[(cur + 16) * pw + k];
                    float ao = sX0[cur * pw + k];
                    float co = sX1[cur * pw + k];
                    win += typ ? (an * an + cn * cn - ao * ao - co * co)
                               : (an + cn - ao - co);
                    cur = p;
                }
                dst[ty * CST + k] = win;
            }
        } else {
            // zero pad columns so WMMA over-reads multiply against clean zeros
            for (int ty = 0; ty < 64; ++ty) dst[ty * CST + k] = 0.0f;
        }
    }
    __syncthreads();

    // ------- horizontal 16-col window sums on the matrix pipe -------
    // Per 16x16 subtile: OUT = A(16x32) x SEL(32x16), SEL banded 0/1,
    // 8 chained V_WMMA_F32_16X16X4_F32 (S and Q interleaved).
    const int wave = tid >> 5;
    const int lane = tid & 31;
    const int mn   = lane & 15;   // A row m / B,D column n
    const int kh   = lane >> 4;   // K-half selector (ISA operand layout)

    float* outc0 = out + (size_t)(b * CC + c0) * HH * WIDTH;
    float* outc1 = out + (size_t)(b * CC + c1) * HH * WIDTH;

    for (int sub = 0; sub < 2; ++sub) {
        int st = wave + 8 * sub;          // 16 subtiles over 8 waves
        int sr = st >> 2, sc = st & 3;

        int w0 = x0 + sc * 16 - 8;
        w0 = w0 < 0 ? 0 : (w0 > 495 ? 495 : w0);
        int kBase = w0 + 1 - pc0;                   // patch-rel window-start col for n=0
        int wn = x0 + sc * 16 + mn - 8;
        wn = wn < 0 ? 0 : (wn > 495 ? 495 : wn);
        int kofs = (wn + 1 - pc0) - kBase;          // 0..15

        const float* rowS = sCS + (sr * 16 + mn) * CST + kBase;
        const float* rowQ = sCQ + (sr * 16 + mn) * CST + kBase;

        v8f accS = {0.f,0.f,0.f,0.f,0.f,0.f,0.f,0.f};
        v8f accQ = {0.f,0.f,0.f,0.f,0.f,0.f,0.f,0.f};
        #pragma unroll
        for (int t = 0; t < 8; ++t) {
            int k0 = t * 4 + kh * 2;                // this lane's K indices: k0, k0+1
            v2f a, aq, bsel;
            a.x  = rowS[k0];  a.y  = rowS[k0 + 1];
            aq.x = rowQ[k0];  aq.y = rowQ[k0 + 1];
            bsel.x = (k0     >= kofs && k0     <= kofs + 15) ? 1.0f : 0.0f;
            bsel.y = (k0 + 1 >= kofs && k0 + 1 <= kofs + 15) ? 1.0f : 0.0f;
            accS = __builtin_amdgcn_wmma_f32_16x16x4_f32(false, a,  false, bsel,
                                                         (short)0, accS, false, false);
            accQ = __builtin_amdgcn_wmma_f32_16x16x4_f32(false, aq, false, bsel,
                                                         (short)0, accQ, false, false);
        }

        // ---------------- normalize + store ----------------
        #pragma unroll
        for (int v = 0; v < 8; ++v) {
            int M  = v + kh * 8;                    // C/D VGPR layout
            int ty = sr * 16 + M;
            int tx = sc * 16 + mn;
            float S = accS[v], Q = accQ[v];
            float mean = S * (1.0f / 512.0f);
            float var  = (Q - S * S * (1.0f / 512.0f)) * (1.0f / 512.0f);
            float rstd = rsqrtf(var + 1e-5f);
            int gy = y0 + ty, gx = x0 + tx;
            float xa = sX0[(gy - pr0) * pw + (gx - pc0)];
            float xb = sX1[(gy - pr0) * pw + (gx - pc0)];
            size_t go = (size_t)gy * WIDTH + gx;
            outc0[go] = (xa - mean) * rstd * ga0 + be0;
            outc1[go] = (xb - mean) * rstd * ga1 + be1;
        }
    }
}

extern "C" void kernel_launch(void* const* d_in, const int* in_sizes, int n_in,
                              void* d_out, int out_size, void* d_ws, size_t ws_size,
                              hipStream_t stream) {
    (void)in_sizes; (void)n_in; (void)d_ws; (void)ws_size; (void)out_size;
    const float* xin   = (const float*)d_in[0];
    const float* gamma = (const float*)d_in[1];
    const float* beta  = (const float*)d_in[2];
    float* outp = (float*)d_out;

    dim3 grid(WIDTH / TILE, HH / TILE, 8 * 16);   // (8, 8, B*G=128)
    dim3 block(256);
    lcn_kernel<<<grid, block, 0, stream>>>(xin, gamma, beta, outp);
}